// SegmenterTorch_91293824843827
// MI455X (gfx1250) — compile-verified
//
#include <hip/hip_runtime.h>
#include <stdint.h>

// WOLA segment->window->overlap-add collapses algebraically to
//   out[b,n] = x[b,n] * G(n),  G from aw*sw with 512-periodic interior + edges.
// Pure streaming kernel: 512 MiB traffic -> ~23us roofline at 23.3 TB/s.
// CDNA5 path exercised: GLOBAL_LOAD_ASYNC_TO_LDS_B128 + s_wait_asynccnt
// double-buffered per-wave pipeline, ds_load_b128 consume, NT b128 stores.

typedef float v4f __attribute__((ext_vector_type(4)));

namespace {
constexpr unsigned kHop          = 512;
constexpr unsigned kNper         = 1u << 22;                 // 4194304 samples/row
constexpr unsigned kBatch        = 16;
constexpr unsigned kVec4Total    = kBatch * (kNper / 4);     // 2^24 float4
constexpr unsigned kItersPerTile = 4;                        // 4 async b128 per lane
constexpr unsigned kTileVec4     = 32 * kItersPerTile;       // 128 float4 = 2KB/tile
constexpr unsigned kNumTiles     = kVec4Total / kTileVec4;   // 131072
constexpr unsigned kBlock        = 256;
constexpr unsigned kWavesPerBlk  = kBlock / 32;              // 8 waves (wave32)
constexpr unsigned kGrid         = 2048;                     // 16384 waves, 8 tiles/wave
}

__global__ __launch_bounds__(kBlock) void wola_stream_kernel(
    const float* __restrict__ x,
    const float* __restrict__ aw,
    const float* __restrict__ sw,
    float* __restrict__ out)
{
    // Gain tables: gt[0]=interior (h[p]+h[p+512]), gt[1]=head (h[p]), gt[2]=tail (h[p+512])
    __shared__ float gt[3][kHop];
    // Per-wave double-buffered staging for async global->LDS copies (2KB per buffer)
    __shared__ float dbuf[kWavesPerBlk][2][kTileVec4 * 4];

    for (unsigned j = threadIdx.x; j < kHop; j += kBlock) {
        float h0 = aw[j] * sw[j];
        float h1 = aw[j + kHop] * sw[j + kHop];
        gt[1][j] = h0;
        gt[2][j] = h1;
        gt[0][j] = h0 + h1;
    }
    __syncthreads();

    const unsigned w      = threadIdx.x >> 5;
    const unsigned lane   = threadIdx.x & 31u;
    const unsigned nWaves = gridDim.x * kWavesPerBlk;
    const unsigned wid    = blockIdx.x * kWavesPerBlk + w;

    const unsigned laneB = lane * 16u;
    // LDS byte offsets (low 32 bits of the flat address are the allocation-relative
    // LDS offset, which is exactly what VDST of the async load expects).
    const unsigned ldsA = (unsigned)(uintptr_t)(&dbuf[w][0][0]) + laneB;
    const unsigned ldsB = (unsigned)(uintptr_t)(&dbuf[w][1][0]) + laneB;
    const unsigned long long xbase = (unsigned long long)(uintptr_t)x;

    // Stage one 2KB tile: 4x global_load_async_to_lds_b128 (GVS mode: s64 base +
    // per-lane u32 byte offset; INST_OFFSET is added on BOTH the global and LDS side).
    auto issue = [&](unsigned ldsOff, unsigned tile) {
        unsigned gb = tile * (kTileVec4 * 16u) + laneB;
        asm volatile("global_load_async_to_lds_b128 %0, %1, %2 offset:0"
                     :: "v"(ldsOff), "v"(gb), "s"(xbase) : "memory");
        asm volatile("global_load_async_to_lds_b128 %0, %1, %2 offset:512"
                     :: "v"(ldsOff), "v"(gb), "s"(xbase) : "memory");
        asm volatile("global_load_async_to_lds_b128 %0, %1, %2 offset:1024"
                     :: "v"(ldsOff), "v"(gb), "s"(xbase) : "memory");
        asm volatile("global_load_async_to_lds_b128 %0, %1, %2 offset:1536"
                     :: "v"(ldsOff), "v"(gb), "s"(xbase) : "memory");
    };

    auto consume = [&](unsigned tile, unsigned bi) {
#pragma unroll
        for (unsigned i = 0; i < kItersPerTile; ++i) {
            unsigned vec = tile * kTileVec4 + i * 32u + lane;
            unsigned e   = vec * 4u;                   // flat element index
            unsigned n0  = e & (kNper - 1u);           // position within row
            unsigned p   = n0 & (kHop - 1u);           // 16B-aligned, no 512 straddle
            unsigned sel = (n0 < kHop) ? 1u : ((n0 >= kNper - kHop) ? 2u : 0u);
            v4f d = *(const v4f*)&dbuf[w][bi][i * 128u + lane * 4u];
            v4f g = *(const v4f*)&gt[sel][p];
            v4f r = d * g;
            __builtin_nontemporal_store(r, (v4f*)(out + e));
        }
    };

    unsigned t  = wid;
    unsigned bi = 0;
    if (t < kNumTiles) issue(ldsA, t);
    while (t < kNumTiles) {
        unsigned tn = t + nWaves;
        if (tn < kNumTiles) {
            issue(bi ? ldsA : ldsB, tn);            // prefetch next tile (other buffer)
            // async loads complete in order: cnt<=4 => current tile's 4 loads landed
            asm volatile("s_wait_asynccnt 4" ::: "memory");
        } else {
            asm volatile("s_wait_asynccnt 0" ::: "memory");
        }
        consume(t, bi);
        bi ^= 1u;
        t = tn;
    }
}

extern "C" void kernel_launch(void* const* d_in, const int* in_sizes, int n_in,
                              void* d_out, int out_size, void* d_ws, size_t ws_size,
                              hipStream_t stream) {
    (void)in_sizes; (void)n_in; (void)out_size; (void)d_ws; (void)ws_size;
    const float* x  = (const float*)d_in[0];
    const float* aw = (const float*)d_in[1];
    const float* sw = (const float*)d_in[2];
    float* out      = (float*)d_out;
    hipLaunchKernelGGL(wola_stream_kernel, dim3(kGrid), dim3(kBlock), 0, stream,
                       x, aw, sw, out);
}